// MinVarianceModel_3762391352067
// MI455X (gfx1250) — compile-verified
//
#include <hip/hip_runtime.h>
#include <hip/hip_bf16.h>

typedef __attribute__((ext_vector_type(2))) float v2f;
typedef __attribute__((ext_vector_type(8))) float v8f;
typedef __attribute__((ext_vector_type(4))) unsigned v4u;
typedef __attribute__((ext_vector_type(8))) int v8i;
typedef __attribute__((ext_vector_type(4))) int v4i;

#define T_SAMPLES 256
#define P_ASSETS  128
#define LDS_STRIDE 132                      // 128 + 4 pad: kills column bank conflicts
#define NTHREADS  256                       // 8 wave32
#define XC_FLOATS (T_SAMPLES * LDS_STRIDE)  // 33792 floats (132 KB)
#define SMEM_FLOATS (XC_FLOATS + P_ASSETS + P_ASSETS + NTHREADS + 8)
#define EPS_W 1e-6f

__device__ __forceinline__ float block_reduce_sum(float v, float* red, int tid) {
  red[tid] = v;
  __syncthreads();
  #pragma unroll
  for (int s = NTHREADS / 2; s > 0; s >>= 1) {
    if (tid < s) red[tid] += red[tid + s];
    __syncthreads();
  }
  float r = red[0];
  __syncthreads();
  return r;
}

extern "C" __global__ __launch_bounds__(NTHREADS)
void minvar_lw_wmma(const float* __restrict__ x,
                    const float* __restrict__ saved_std,
                    float* __restrict__ out)
{
  extern __shared__ float smem[];
  float* Xc   = smem;                 // [256][132] raw -> centered -> (reused) gram[128][132]
  float* mx   = Xc + XC_FLOATS;       // [128] column means, later reused as solve vector y
  float* sstd = mx + P_ASSETS;        // [128]
  float* red  = sstd + P_ASSETS;      // [256] reduction scratch
  float* scal = red + NTHREADS;       // [8] broadcast scalars

  const int tid = threadIdx.x;
  const int b   = blockIdx.x;
  const float* xb = x + (size_t)b * (T_SAMPLES * P_ASSETS);

  // ---- phase 1: TDM DMA of the whole 256x128 f32 tile into LDS ----
  // D# pads the LDS destination natively: pad_interval=128 DWORDs (code 6),
  // pad_amount=4 DWORDs (code 3) => rows land at stride 132 floats.
  if (tid < 32) {                       // wave 0 only (TDM ignores EXEC; avoid 8x duplicate DMA)
    const unsigned long long gaddr = (unsigned long long)(uintptr_t)xb;
    const unsigned lds_off = (unsigned)(uintptr_t)Xc;   // low 32 bits = LDS byte offset

    v4u g0;
    g0[0] = 1u;                                          // count=1, no gather
    g0[1] = lds_off;                                     // lds_addr (bytes)
    g0[2] = (unsigned)(gaddr & 0xFFFFFFFFull);           // global_addr[31:0]
    g0[3] = (unsigned)((gaddr >> 32) & 0x01FFFFFFull)    // global_addr[56:32]
            | 0x80000000u;                               // type=2 ("image")

    v8i g1;
    g1[0] = 0x07920000;            // data_size=4B | pad_enable | pad_interval=6 | pad_amount=3
    g1[1] = (int)(128u << 16);     // tensor_dim0   = 128 elements/row   (bits[63:48])
    g1[2] = (int)(256u << 16);     // tensor_dim1   = 256 rows           (bits[95:80])
    g1[3] = (int)(128u << 16);     // tile_dim0     = 128                (bits[127:112])
    g1[4] = (int)256;              // tile_dim1     = 256                (bits[143:128])
    g1[5] = (int)128;              // tensor_dim0_stride = 128 elements  (bits[207:160])
    g1[6] = 0;
    g1[7] = 0;

    v4i gz4 = {0, 0, 0, 0};                  // 2-D tensor: groups 2/3 unused
    v8i gz8 = {0, 0, 0, 0, 0, 0, 0, 0};      // unused trailing group (6-arg toolchain form)

    __builtin_amdgcn_tensor_load_to_lds(g0, g1, gz4, gz4, gz8, 0);
    __builtin_amdgcn_s_wait_tensorcnt(0);
  }
  if (tid < P_ASSETS) sstd[tid] = saved_std[tid];
  __syncthreads();

  // ---- phase 2: column means of raw x (additive saved_mean cancels in centering) ----
  if (tid < P_ASSETS) {
    float s = 0.f;
    for (int r = 0; r < T_SAMPLES; ++r) s += Xc[r * LDS_STRIDE + tid];
    mx[tid] = s * (1.0f / T_SAMPLES);
  }
  __syncthreads();

  // ---- phase 3: center + scale in place: Xc = std[p] * (x - colmean[p]) ----
  for (int i = tid; i < (T_SAMPLES * P_ASSETS) / 4; i += NTHREADS) {
    int idx = i << 2;
    int r = idx >> 7;
    int c = idx & 127;
    float4 v = *(float4*)&Xc[r * LDS_STRIDE + c];
    v.x = sstd[c + 0] * (v.x - mx[c + 0]);
    v.y = sstd[c + 1] * (v.y - mx[c + 1]);
    v.z = sstd[c + 2] * (v.z - mx[c + 2]);
    v.w = sstd[c + 3] * (v.w - mx[c + 3]);
    *(float4*)&Xc[r * LDS_STRIDE + c] = v;
  }
  __syncthreads();

  // ---- phase 4: beta_raw = sum_n (sum_p Xc[n,p]^2)^2  (one row per thread, T==NTHREADS) ----
  float beta_raw;
  {
    const float* rowp = &Xc[tid * LDS_STRIDE];
    float rs = 0.f;
    for (int c = 0; c < P_ASSETS; ++c) { float v = rowp[c]; rs += v * v; }
    beta_raw = block_reduce_sum(rs * rs, red, tid);
  }

  // ---- phase 5: gram = Xc^T Xc via V_WMMA_F32_16X16X4_F32 ----
  // wave w owns gram rows [16w, 16w+16); 8 output tiles of 16x16; K-loop over 256 in steps of 4.
  const int lane  = tid & 31;
  const int wave  = tid >> 5;
  const int lo16  = lane & 15;
  const int khalf = lane >> 4;           // A frag: lanes 0-15 -> K={k0,k0+1}, 16-31 -> K={k0+2,k0+3}
  const int mrow  = wave * 16 + lo16;

  v8f acc[8] = {};

  for (int k0 = 0; k0 < T_SAMPLES; k0 += 4) {
    const int ka = k0 + khalf * 2;
    v2f afrag;                                            // A[m][k] = Xc[k][m]
    afrag[0] = Xc[(ka + 0) * LDS_STRIDE + mrow];
    afrag[1] = Xc[(ka + 1) * LDS_STRIDE + mrow];
    #pragma unroll
    for (int j = 0; j < 8; ++j) {
      const int ncol = j * 16 + lo16;
      v2f bfrag;                                          // B[k][n] = Xc[k][n]
      bfrag[0] = Xc[(ka + 0) * LDS_STRIDE + ncol];
      bfrag[1] = Xc[(ka + 1) * LDS_STRIDE + ncol];
      // 8 args: (neg_a, A, neg_b, B, c_mod, C, reuse_a, reuse_b)
      acc[j] = __builtin_amdgcn_wmma_f32_16x16x4_f32(
          false, afrag, false, bfrag, (short)0, acc[j], false, false);
    }
  }
  __syncthreads();   // all LDS reads of Xc done; safe to overwrite with gram

  // write gram over Xc region. C layout: VGPR r -> (M=r, lanes 0-15) / (M=8+r, lanes 16-31)
  float* gram = Xc;
  {
    const int rofs = (lane >= 16) ? 8 : 0;
    #pragma unroll
    for (int r = 0; r < 8; ++r) {
      const int row = wave * 16 + r + rofs;
      #pragma unroll
      for (int j = 0; j < 8; ++j)
        gram[row * LDS_STRIDE + j * 16 + lo16] = acc[j][r];
    }
  }
  __syncthreads();

  // ---- phase 6: shrinkage scalars ----
  float trg = block_reduce_sum((tid < P_ASSETS) ? gram[tid * LDS_STRIDE + tid] : 0.f,
                               red, tid);
  float fro_part = 0.f;
  for (int idx = tid; idx < P_ASSETS * P_ASSETS; idx += NTHREADS) {
    int r = idx >> 7, c = idx & 127;
    float g = gram[r * LDS_STRIDE + c];
    fro_part += g * g;
  }
  float fro = block_reduce_sum(fro_part, red, tid);

  if (tid == 0) {
    const float n = (float)T_SAMPLES, p = (float)P_ASSETS;
    float delta_    = fro / (n * n);
    float trace_emp = trg / n;
    float mu        = trace_emp / p;
    float beta_     = (beta_raw / n - delta_) / (p * n);
    float delta_den = (delta_ - 2.f * mu * trace_emp + p * mu * mu) / p;
    float bmin      = fminf(beta_, delta_den);
    float shrink    = (delta_den > 0.f) ? (bmin / delta_den) : 0.f;
    scal[0] = (1.f - shrink) / n;   // scale on gram  -> (1-s)*emp_cov
    scal[1] = shrink * mu;          // diagonal add   -> s*mu*I
  }
  __syncthreads();
  const float acov = scal[0], dcov = scal[1];

  // ---- phase 7: cov in place ----
  for (int idx = tid; idx < P_ASSETS * P_ASSETS; idx += NTHREADS) {
    int r = idx >> 7, c = idx & 127;
    float g = gram[r * LDS_STRIDE + c];
    gram[r * LDS_STRIDE + c] = acov * g + ((r == c) ? dcov : 0.f);
  }
  __syncthreads();

  // ---- phase 8: cooperative Cholesky (lower) in LDS ----
  float* A = gram;
  for (int k = 0; k < P_ASSETS; ++k) {
    if (tid == 0) A[k * LDS_STRIDE + k] = sqrtf(fmaxf(A[k * LDS_STRIDE + k], 1e-30f));
    __syncthreads();
    const float dk = A[k * LDS_STRIDE + k];
    if (tid > k && tid < P_ASSETS) A[tid * LDS_STRIDE + k] /= dk;
    __syncthreads();
    if (tid > k && tid < P_ASSETS) {
      const float lik = A[tid * LDS_STRIDE + k];
      for (int j = k + 1; j <= tid; ++j)
        A[tid * LDS_STRIDE + j] -= lik * A[j * LDS_STRIDE + k];
    }
    __syncthreads();
  }

  // ---- phase 9: solve L y = 1, then L^T z = y (column-oriented; y reuses mx) ----
  float* y = mx;
  if (tid < P_ASSETS) y[tid] = 1.0f;
  __syncthreads();
  for (int j = 0; j < P_ASSETS; ++j) {
    if (tid == 0) y[j] /= A[j * LDS_STRIDE + j];
    __syncthreads();
    const float yj = y[j];
    if (tid > j && tid < P_ASSETS) y[tid] -= A[tid * LDS_STRIDE + j] * yj;
    __syncthreads();
  }
  for (int j = P_ASSETS - 1; j >= 0; --j) {
    if (tid == 0) y[j] /= A[j * LDS_STRIDE + j];
    __syncthreads();
    const float zj = y[j];
    if (tid < j) y[tid] -= A[j * LDS_STRIDE + tid] * zj;   // L^T[i][j] = L[j][i]
    __syncthreads();
  }

  // ---- phase 10: clip + normalize ----
  const float wv  = (tid < P_ASSETS) ? fmaxf(y[tid], EPS_W) : 0.f;
  const float tot = block_reduce_sum(wv, red, tid);
  if (tid < P_ASSETS) out[(size_t)b * P_ASSETS + tid] = wv / tot;
}

extern "C" void kernel_launch(void* const* d_in, const int* in_sizes, int n_in,
                              void* d_out, int out_size, void* d_ws, size_t ws_size,
                              hipStream_t stream) {
  const float* x         = (const float*)d_in[0];
  const float* saved_std = (const float*)d_in[2];   // saved_mean (d_in[1]) cancels in centering
  float* out = (float*)d_out;

  const int B = in_sizes[0] / (T_SAMPLES * P_ASSETS);
  const size_t smemBytes = (size_t)SMEM_FLOATS * sizeof(float);   // ~137 KB of the 320 KB WGP LDS

  (void)hipFuncSetAttribute((const void*)minvar_lw_wmma,
                            hipFuncAttributeMaxDynamicSharedMemorySize, (int)smemBytes);

  minvar_lw_wmma<<<dim3(B), dim3(NTHREADS), smemBytes, stream>>>(x, saved_std, out);

  (void)n_in; (void)d_ws; (void)ws_size; (void)out_size;
}